// KMeans_14353780703859
// MI455X (gfx1250) — compile-verified
//
#include <hip/hip_runtime.h>
#include <math.h>
#include <stdint.h>

#define B_    8
#define N_    65536
#define D_    32
#define K_    5
#define ITERS 100
#define PPB   1024              // points per block
#define NWAVES 8                // 256 threads / wave32
#define TPW   (PPB / (NWAVES * 16))  // tiles of 16 points per wave = 8

typedef __attribute__((ext_vector_type(2))) float v2f;
typedef __attribute__((ext_vector_type(8))) float v8f;

// Order-preserving float -> uint map (total order, matches < on floats).
__device__ __forceinline__ uint32_t f32_ord(float f) {
    uint32_t u = __float_as_uint(f);
    return (u & 0x80000000u) ? ~u : (u | 0x80000000u);
}

// ---------------------------------------------------------------------------
// Init: pick 5 deterministic seed points as initial centers, zero accumulators.
// ---------------------------------------------------------------------------
__global__ void kmeans_init(const float* __restrict__ X,
                            float* __restrict__ centers,
                            float* __restrict__ sums,
                            float* __restrict__ counts) {
    const int b = blockIdx.x;
    const int tid = threadIdx.x;
    const int seed[K_] = {11, 13107, 26214, 39321, 52428};
    if (tid < K_ * D_) {
        int k = tid / D_, d = tid % D_;
        centers[b * K_ * D_ + tid] = X[((size_t)b * N_ + seed[k]) * D_ + d];
        sums[b * K_ * D_ + tid] = 0.0f;
    }
    if (tid < K_) counts[b * K_ + tid] = 0.0f;
}

// ---------------------------------------------------------------------------
// Assign: per-wave 16x16 WMMA tiles of X . centers^T, branchless packed-key
// argmin over centers, LDS-accumulated partial sums/counts, one global atomic
// per (k,d) per block.
// ---------------------------------------------------------------------------
__global__ void __launch_bounds__(256)
kmeans_assign(const float* __restrict__ X,
              const float* __restrict__ centersG,
              float* __restrict__ sumsG,
              float* __restrict__ countsG,
              float* __restrict__ labelsOut,
              int writeLabels) {
    __shared__ float ldsC[K_ * D_];
    __shared__ float ldsC2[16];
    __shared__ float ldsSums[K_ * D_];
    __shared__ float ldsCnt[K_];

    const int b   = blockIdx.y;
    const int tid = threadIdx.x;

    if (tid < K_ * D_) {
        ldsC[tid] = centersG[b * K_ * D_ + tid];
        ldsSums[tid] = 0.0f;
    }
    if (tid < K_) ldsCnt[tid] = 0.0f;
    __syncthreads();

    if (tid < 16) {
        float s = INFINITY;                 // pad centers 5..15 out of argmin
        if (tid < K_) {
            s = 0.0f;
            for (int d = 0; d < D_; ++d) { float c = ldsC[tid * D_ + d]; s += c * c; }
        }
        ldsC2[tid] = s;
    }
    __syncthreads();

    const int lane = tid & 31;
    const int wave = tid >> 5;
    const int n16  = lane & 15;
    const int half = lane >> 4;

    // B fragments: centers^T (4x16 per chunk), columns >= K_ zeroed. Invariant.
    v2f bf[8];
#pragma unroll
    for (int c = 0; c < 8; ++c) {
        if (n16 < K_) {
            bf[c].x = ldsC[n16 * D_ + 4 * c + 2 * half];
            bf[c].y = ldsC[n16 * D_ + 4 * c + 2 * half + 1];
        } else {
            bf[c].x = 0.0f; bf[c].y = 0.0f;
        }
    }
    const float c2reg = ldsC2[n16];

    const int blockStart = blockIdx.x * PPB;

    for (int t = 0; t < TPW; ++t) {
        const int pbase = blockStart + (wave * TPW + t) * 16;
        const float* Xp = X + ((size_t)b * N_ + pbase) * D_;

        // 16x16 f32 accumulator: dot[m][k] = sum_d X[pbase+m][d] * C[k][d]
        v8f acc = {};
#pragma unroll
        for (int c = 0; c < 8; ++c) {
            v2f a;
            const float* ap = Xp + (size_t)n16 * D_ + 4 * c + 2 * half;
            a.x = ap[0];
            a.y = ap[1];
            acc = __builtin_amdgcn_wmma_f32_16x16x4_f32(
                false, a, false, bf[c], (short)0, acc, false, false);
        }

        // Preload the 16 points' value at dim=lane (coalesced, single base).
        float xv[16];
        const float* Xw = Xp + lane;
#pragma unroll
        for (int r = 0; r < 16; ++r) xv[r] = Xw[r * D_];

#pragma unroll
        for (int r = 0; r < 8; ++r) {
            // lane holds (c2[k] - 2*dot) for point m = r + 8*half, k = lane&15.
            // Pack into order-preserving 64-bit key with index in LSBs:
            // branchless min reduction + exact lowest-index tie-break.
            float v = c2reg - 2.0f * acc[r];
            uint64_t key = ((uint64_t)f32_ord(v) << 5) | (uint32_t)n16;
            // Valid centers live in lanes 0..4 (and 16..20): xor {4,2,1} covers them.
#pragma unroll
            for (int off = 4; off >= 1; off >>= 1) {
                uint32_t lo  = (uint32_t)key;
                uint32_t hi  = (uint32_t)(key >> 32);
                uint32_t olo = (uint32_t)__shfl_xor((int)lo, off, 32);
                uint32_t ohi = (uint32_t)__shfl_xor((int)hi, off, 32);
                uint64_t ok  = ((uint64_t)ohi << 32) | olo;
                if (ok < key) key = ok;     // v_cmp_lt_u64 + cndmask, no branches
            }
            const int idx  = (int)(key & 31u);
            const int lab0 = __builtin_amdgcn_readlane(idx, 0);   // point pbase+r
            const int lab1 = __builtin_amdgcn_readlane(idx, 16);  // point pbase+r+8

            atomicAdd(&ldsSums[lab0 * D_ + lane], xv[r]);
            atomicAdd(&ldsSums[lab1 * D_ + lane], xv[r + 8]);
            if (lane == 0) {
                atomicAdd(&ldsCnt[lab0], 1.0f);
                atomicAdd(&ldsCnt[lab1], 1.0f);
                if (writeLabels) {
                    labelsOut[(size_t)b * N_ + pbase + r]     = (float)lab0;
                    labelsOut[(size_t)b * N_ + pbase + r + 8] = (float)lab1;
                }
            }
        }
    }

    __syncthreads();
    if (tid < K_ * D_) atomicAdd(&sumsG[b * K_ * D_ + tid], ldsSums[tid]);
    if (tid < K_)      atomicAdd(&countsG[b * K_ + tid],    ldsCnt[tid]);
}

// ---------------------------------------------------------------------------
// Update: centers = sums / counts; zero accumulators; final iter also writes
// centers + percentages to d_out.
// ---------------------------------------------------------------------------
__global__ void kmeans_update(float* __restrict__ centersG,
                              float* __restrict__ sumsG,
                              float* __restrict__ countsG,
                              float* __restrict__ outCenters,
                              float* __restrict__ outPct,
                              int finalFlag) {
    const int b = blockIdx.x, tid = threadIdx.x;
    if (tid < K_ * D_) {
        int k = tid / D_;
        float cnt = countsG[b * K_ + k];
        float nc  = sumsG[b * K_ * D_ + tid] / cnt;
        centersG[b * K_ * D_ + tid] = nc;
        if (finalFlag) outCenters[b * K_ * D_ + tid] = nc;
    }
    __syncthreads();
    if (tid < K_ * D_) sumsG[b * K_ * D_ + tid] = 0.0f;
    if (tid < K_) {
        if (finalFlag) outPct[b * K_ + tid] = countsG[b * K_ + tid] / (float)N_;
        countsG[b * K_ + tid] = 0.0f;
    }
}

// ---------------------------------------------------------------------------
extern "C" void kernel_launch(void* const* d_in, const int* in_sizes, int n_in,
                              void* d_out, int out_size, void* d_ws, size_t ws_size,
                              hipStream_t stream) {
    const float* X = (const float*)d_in[0];

    // Workspace: centers | sums | counts (all f32)
    float* centers = (float*)d_ws;
    float* sums    = centers + B_ * K_ * D_;
    float* counts  = sums    + B_ * K_ * D_;

    // Output: centers (1280) | labels (524288, stored as float) | pct (40)
    float* out        = (float*)d_out;
    float* outCenters = out;
    float* outLabels  = out + B_ * K_ * D_;
    float* outPct     = outLabels + (size_t)B_ * N_;

    kmeans_init<<<B_, 192, 0, stream>>>(X, centers, sums, counts);

    for (int it = 0; it < ITERS; ++it) {
        const int last = (it == ITERS - 1) ? 1 : 0;
        kmeans_assign<<<dim3(N_ / PPB, B_), 256, 0, stream>>>(
            X, centers, sums, counts, outLabels, last);
        kmeans_update<<<B_, 192, 0, stream>>>(
            centers, sums, counts, outCenters, outPct, last);
    }
}